// SelfAttention_48936857370889
// MI455X (gfx1250) — compile-verified
//
#include <hip/hip_runtime.h>
#include <math.h>

typedef __attribute__((ext_vector_type(2))) float v2f;
typedef __attribute__((ext_vector_type(4))) float v4f;
typedef __attribute__((ext_vector_type(8))) float v8f;

#define EMB 256
#define HH  64
#define WW  64
#define BB  8
#define NN  4096   // HH*WW

#if defined(__gfx1250__) && __has_builtin(__builtin_amdgcn_global_load_async_to_lds_b128)
#define USE_ASYNC_LDS 1
#else
#define USE_ASYNC_LDS 0
#endif

#if USE_ASYNC_LDS
// builtin signature (from hipcc diagnostic): arg0 is AS(1) pointer to
// 'int __attribute__((vector_size(16)))'; arg1 is the LDS-side pointer.
typedef int v4i_vs __attribute__((vector_size(16)));
typedef __attribute__((address_space(1))) v4i_vs* g_v4i_p;
typedef __attribute__((address_space(3))) v4i_vs* l_v4i_p;
#endif

#if defined(__gfx1250__)
static __device__ __forceinline__ void wait_async0() {
#if __has_builtin(__builtin_amdgcn_s_wait_asynccnt)
    __builtin_amdgcn_s_wait_asynccnt(0);
#else
    asm volatile("s_wait_asynccnt 0x0" ::: "memory");
#endif
}
#endif

// ---- half-wave (16-lane) reductions: rows of a 16x16 WMMA C-tile live in
// ---- lanes 0..15 (M=0..7) and 16..31 (M=8..15); xor masks 1,2,4,8 stay
// ---- inside each half of the wave32.
static __device__ __forceinline__ float half_reduce_max(float v) {
#pragma unroll
    for (int m = 1; m <= 8; m <<= 1) v = fmaxf(v, __shfl_xor(v, m, 32));
    return v;
}
static __device__ __forceinline__ float half_reduce_sum(float v) {
#pragma unroll
    for (int m = 1; m <= 8; m <<= 1) v += __shfl_xor(v, m, 32);
    return v;
}

// =====================================================================
// Kernel 0: repack conv weights [tap][k][oc] -> exact B-fragment order:
// wf[(((tap*16+ot)*64)+j)*32 + lane] = { w[tap][4j+2hi][ot*16+lo],
//                                        w[tap][4j+2hi+1][ot*16+lo] }
// so each WMMA B-fragment is a single contiguous b64 load per lane.
// =====================================================================
__global__ __launch_bounds__(256)
void repack_w(const float* __restrict__ w, v2f* __restrict__ wf)
{
    const int idx = blockIdx.x * 256 + threadIdx.x;   // 0 .. 294911
    const int lane = idx & 31;
    const int j   = (idx >> 5) & 63;
    const int ot  = (idx >> 11) & 15;
    const int tap = idx >> 15;                        // 0..8
    const int hi = lane >> 4, lo = lane & 15;
    const int k  = 4 * j + 2 * hi;
    const int oc = ot * 16 + lo;
    v2f v;
    v.x = w[((size_t)tap * EMB + k) * EMB + oc];
    v.y = w[((size_t)tap * EMB + k + 1) * EMB + oc];
    wf[idx] = v;
}

// =====================================================================
// Kernel 1: 3x3 same-padded conv as implicit GEMM with fp32 WMMA.
// One wave per (16-pixel strip, 16-output-channel tile).
// Boundary handling via clamped address + 0/1 mask (no exec divergence).
// Two interleaved accumulators break the WMMA dependency chain.
// =====================================================================
__global__ __launch_bounds__(32)
void conv3x3_wmma(const float* __restrict__ in, const v2f* __restrict__ wf,
                  const float* __restrict__ bias, float* __restrict__ x1)
{
    const int lane = threadIdx.x;
    const int hi = lane >> 4;          // 0: K={0,1}, 1: K={2,3} (A/B layout)
    const int lo = lane & 15;
    const int strip = blockIdx.x;      // 0..2047
    const int ot  = blockIdx.y;        // oc tile 0..15
    const int b  = strip >> 8;
    const int n0 = (strip & 255) << 4; // first pixel of strip
    const int y  = n0 >> 6;            // 16 | 64 => whole strip on one row
    const int x0 = n0 & 63;

    v8f acc0 = {}, acc1 = {};
    const float* inb = in + (size_t)b * NN * EMB;

    for (int dy = -1; dy <= 1; ++dy) {
        const int yy  = y + dy;
        const int yc  = min(max(yy, 0), HH - 1);
        const bool yok = (yy >= 0) & (yy < HH);
        for (int dx = -1; dx <= 1; ++dx) {
            const int xraw = x0 + lo + dx;            // this lane's A row pixel
            const int xc   = min(max(xraw, 0), WW - 1);
            const bool ok  = yok & (xraw >= 0) & (xraw < WW);
            const float msk = ok ? 1.0f : 0.0f;
            const float* ap = inb + ((size_t)(yc * WW + xc)) * EMB + 2 * hi;
            const int tap = (dy + 1) * 3 + (dx + 1);
            const v2f* wp = wf + (size_t)((tap * 16 + ot) * 64) * 32 + lane;

            for (int jb = 0; jb < 64; jb += 8) {      // K = 256 in steps of 4
                v2f a[8], bf[8];
#pragma unroll
                for (int u = 0; u < 8; ++u) {
                    a[u] = *(const v2f*)(ap + 4 * (jb + u));
                    a[u].x *= msk; a[u].y *= msk;
                    bf[u] = wp[(size_t)(jb + u) * 32];
                }
#pragma unroll
                for (int u = 0; u < 8; ++u) {
                    if (u & 1)
                        acc1 = __builtin_amdgcn_wmma_f32_16x16x4_f32(
                                   false, a[u], false, bf[u], (short)0, acc1, false, false);
                    else
                        acc0 = __builtin_amdgcn_wmma_f32_16x16x4_f32(
                                   false, a[u], false, bf[u], (short)0, acc0, false, false);
                }
            }
        }
    }

    const float bv = bias[ot * 16 + lo];
    float* outp = x1 + ((size_t)b * NN + n0) * EMB + ot * 16 + lo;
#pragma unroll
    for (int r = 0; r < 8; ++r) {
        const int row = r + 8 * hi;                  // C/D layout row mapping
        outp[(size_t)row * EMB] = acc0[r] + acc1[r] + bv;
    }
}

// =====================================================================
// Kernel 2: flash-style self-attention + fused output multiply.
// Block = 128 threads (4 waves). Each wave: one 16-row query tile.
// 16x256 K/V tile staged in LDS via async global->LDS (ASYNCcnt path),
// shared by the 4 waves. Online softmax via half-wave shuffles.
// =====================================================================
__global__ __launch_bounds__(128)
void attn_wmma(const float* __restrict__ x1, const float* __restrict__ in,
               float* __restrict__ out)
{
    __shared__ float Xt[16 * EMB];      // 16 KB key/value tile (row-major)
    __shared__ float Pl[4][16 * 16];    // per-wave P tile (C-layout -> A-layout)

    const int tid  = threadIdx.x;
    const int lane = tid & 31;
    const int wv   = tid >> 5;
    const int hi = lane >> 4, lo = lane & 15;
    const int b  = blockIdx.x >> 6;
    const int r0 = ((blockIdx.x & 63) << 6) + (wv << 4);   // query row base

    const float* xb = x1 + (size_t)b * NN * EMB;

    v8f o[16];                                   // 16 col-tiles x v8f = 256 ch
#pragma unroll
    for (int c = 0; c < 16; ++c) o[c] = (v8f){};
    float mrow[8], lrow[8];
#pragma unroll
    for (int r = 0; r < 8; ++r) { mrow[r] = -3.0e38f; lrow[r] = 0.f; }

    // per-lane Q fragment pointer: row r0+lo, channel pair base 2*hi
    const float* qp = xb + (size_t)(r0 + lo) * EMB + 2 * hi;

    for (int mt = 0; mt < 256; ++mt) {
        const int m0 = mt << 4;
        __syncthreads();                         // protect Xt overwrite
        {   // load X tile [16][EMB]: 1024 float4 across 128 threads
            const float* src = xb + (size_t)m0 * EMB;
#if USE_ASYNC_LDS
#pragma unroll
            for (int it = 0; it < 8; ++it) {
                const int i4 = tid + it * 128;
                __builtin_amdgcn_global_load_async_to_lds_b128(
                    (g_v4i_p)(src + i4 * 4),
                    (l_v4i_p)&Xt[i4 * 4],
                    0, 0);
            }
            wait_async0();
#else
#pragma unroll
            for (int it = 0; it < 8; ++it) {
                const int i4 = tid + it * 128;
                *(v4f*)&Xt[i4 * 4] = *(const v4f*)&src[i4 * 4];
            }
#endif
        }
        if (mt + 1 < 256)                        // pull next tile toward L2/L0
            __builtin_prefetch(xb + (size_t)(m0 + 16) * EMB + tid * 32, 0, 0);
        __syncthreads();

        // ---- S = Q * K^T (16x16), K rows = Xt rows, K-dim = 256 channels.
        // Batched fragment loads + two accumulator chains.
        v8f s0 = {}, s1 = {};
        for (int jb = 0; jb < 64; jb += 8) {
            v2f qa[8], kb[8];
#pragma unroll
            for (int u = 0; u < 8; ++u) {
                qa[u] = *(const v2f*)(qp + 4 * (jb + u));
                kb[u] = *(const v2f*)&Xt[lo * EMB + 4 * (jb + u) + 2 * hi];
            }
#pragma unroll
            for (int u = 0; u < 8; ++u) {
                if (u & 1)
                    s1 = __builtin_amdgcn_wmma_f32_16x16x4_f32(
                             false, qa[u], false, kb[u], (short)0, s1, false, false);
                else
                    s0 = __builtin_amdgcn_wmma_f32_16x16x4_f32(
                             false, qa[u], false, kb[u], (short)0, s0, false, false);
            }
        }

        // ---- online softmax (per row, rows split across lane halves)
        float scl[8];
#pragma unroll
        for (int r = 0; r < 8; ++r) {
            const float sv   = s0[r] + s1[r];
            const float tmax = half_reduce_max(sv);
            const float mnew = fmaxf(mrow[r], tmax);
            const float sc   = __expf(mrow[r] - mnew);
            const float p    = __expf(sv - mnew);
            lrow[r] = lrow[r] * sc + half_reduce_sum(p);
            mrow[r] = mnew;
            scl[r]  = sc;
            Pl[wv][(r + 8 * hi) * 16 + lo] = p;  // row-major P tile
        }
#pragma unroll
        for (int c = 0; c < 16; ++c)
#pragma unroll
            for (int r = 0; r < 8; ++r) o[c][r] *= scl[r];

        __syncthreads();                         // P visible; Xt still live

        // ---- P as A-fragments (K = 16 split into 4 x K=4)
        v2f pa[4];
#pragma unroll
        for (int j = 0; j < 4; ++j)
            pa[j] = *(const v2f*)&Pl[wv][lo * 16 + 4 * j + 2 * hi];

        // ---- O += P @ V, V = Xt (16 independent accumulator chains)
#pragma unroll
        for (int c = 0; c < 16; ++c) {
#pragma unroll
            for (int j = 0; j < 4; ++j) {
                const int kk = 4 * j + 2 * hi;
                v2f bf;
                bf.x = Xt[kk * EMB + c * 16 + lo];
                bf.y = Xt[(kk + 1) * EMB + c * 16 + lo];
                o[c] = __builtin_amdgcn_wmma_f32_16x16x4_f32(
                           false, pa[j], false, bf, (short)0, o[c], false, false);
            }
        }
    }

    // ---- epilogue: softmax normalize, /sqrt(E)=16, multiply by inputs.
    // (b,w,h,e) reshape is a memory no-op since H==W -> flat elementwise.
    const float* ib = in + (size_t)b * NN * EMB;
    float* ob = out + (size_t)b * NN * EMB;
    float inv[8];
#pragma unroll
    for (int r = 0; r < 8; ++r) inv[r] = 1.0f / (lrow[r] * 16.0f);
#pragma unroll
    for (int c = 0; c < 16; ++c) {
        const int ch = c * 16 + lo;
#pragma unroll
        for (int r = 0; r < 8; ++r) {
            const size_t idx = (size_t)(r0 + r + 8 * hi) * EMB + ch;
            ob[idx] = o[c][r] * inv[r] * ib[idx];
        }
    }
}

extern "C" void kernel_launch(void* const* d_in, const int* in_sizes, int n_in,
                              void* d_out, int out_size, void* d_ws, size_t ws_size,
                              hipStream_t stream)
{
    const float* in = (const float*)d_in[0];   // [8,64,64,256]
    const float* cw = (const float*)d_in[1];   // [3,3,256,256] HWIO
    const float* cb = (const float*)d_in[2];   // [256]
    float* out = (float*)d_out;                // [8,64,64,256]
    float* x1  = (float*)d_ws;                 // conv output, 33.5 MB
    v2f*   wf  = (v2f*)(x1 + (size_t)BB * NN * EMB);  // repacked weights, 2.36 MB

    repack_w<<<1152, 256, 0, stream>>>(cw, wf);

    dim3 g1(2048, 16);                         // strips x oc-tiles
    conv3x3_wmma<<<g1, 32, 0, stream>>>(in, wf, cb, x1);
    attn_wmma<<<512, 128, 0, stream>>>(x1, in, out);
}